// GMM_44332652429390
// MI455X (gfx1250) — compile-verified
//
#include <hip/hip_runtime.h>
#include <hip/hip_bf16.h>
#include <math.h>

typedef __attribute__((ext_vector_type(2))) float v2f;
typedef __attribute__((ext_vector_type(8))) float v8f;

#define NMIX 64
#define NDIM 16
#define PI2F 6.283185307f

// ---------------- Precompute P = 1/cov, Q = -2*mean/cov, s[m] = log(w*c*cdet) - 0.5*r[m]
__global__ void gmm_precompute(const float* __restrict__ wghts,
                               const float* __restrict__ means,
                               const float* __restrict__ dcovs,
                               float* __restrict__ Pws,
                               float* __restrict__ Qws,
                               float* __restrict__ Sws) {
    int m = threadIdx.x;
    if (m >= NMIX) return;
    float sum_log = 0.0f, sum_pm = 0.0f;
    for (int d = 0; d < NDIM; ++d) {
        float cov = dcovs[m * NDIM + d];
        float p   = 1.0f / cov;
        float mu  = means[m * NDIM + d];
        Pws[m * NDIM + d] = p;
        Qws[m * NDIM + d] = -2.0f * p * mu;
        sum_log += logf(cov);
        sum_pm  += p * mu * mu;
    }
    // prob = exp(-0.5*G + s), G = x^2.P + x.Q
    Sws[m] = logf(wghts[m]) - 0.5f * (float)NDIM * logf(PI2F)
             - 0.5f * sum_log - 0.5f * sum_pm;
}

// ---------------- Main: per-wave 16 t-rows, WMMA f32 16x16x4 for both contractions
__global__ __launch_bounds__(256) void gmm_main(
        const float* __restrict__ data,
        const float* __restrict__ means,
        const float* __restrict__ Pws,
        const float* __restrict__ Qws,
        const float* __restrict__ Sws,
        float* __restrict__ e_means,
        float* __restrict__ partials) {
    __shared__ float lds_post[8 * 16 * 64];   // per-wave 16x64 posterior tile (32 KB)
    __shared__ float wave_sums[8];

    const int tid  = threadIdx.x;
    const int wave = tid >> 5;
    const int lane = tid & 31;
    const int g    = lane >> 4;    // half-wave (K split)
    const int n    = lane & 15;    // column / row-in-tile index
    const int t_base = (blockIdx.x * 8 + wave) * 16;

    // A fragments (rows = t, K = d): lane n = row; K = k + 2*g + {0,1}
    v2f aF[4], a2F[4];
#pragma unroll
    for (int kk = 0; kk < 4; ++kk) {
        const int k = kk * 4;
        const float x = data[(t_base + n) * NDIM + k + 2 * g];
        const float y = data[(t_base + n) * NDIM + k + 1 + 2 * g];
        aF[kk].x = x;      aF[kk].y = y;
        a2F[kk].x = x * x; a2F[kk].y = y * y;
    }

    // Stage 1: G[t,m] = data^2 @ P^T + data @ Q^T  (4 m-tiles, K=16 each term)
    v8f acc[4];
#pragma unroll
    for (int mt = 0; mt < 4; ++mt) {
        v8f c = {};
        const int mrow = (mt * 16 + n) * NDIM;
#pragma unroll
        for (int kk = 0; kk < 4; ++kk) {
            const int k = kk * 4;
            v2f bP, bQ;
            bP.x = Pws[mrow + k + 2 * g];
            bP.y = Pws[mrow + k + 1 + 2 * g];
            bQ.x = Qws[mrow + k + 2 * g];
            bQ.y = Qws[mrow + k + 1 + 2 * g];
            c = __builtin_amdgcn_wmma_f32_16x16x4_f32(false, a2F[kk], false, bP,
                                                      (short)0, c, false, false);
            c = __builtin_amdgcn_wmma_f32_16x16x4_f32(false, aF[kk], false, bQ,
                                                      (short)0, c, false, false);
        }
        acc[mt] = c;
    }

    // Stage 2: probs = exp(-0.5*G + s[m]); likes = sum_m; posteriors -> LDS
    float s_m[4];
#pragma unroll
    for (int mt = 0; mt < 4; ++mt) s_m[mt] = Sws[mt * 16 + n];

    float ll_acc = 0.0f;
    float* myLds = lds_post + wave * (16 * 64);
#pragma unroll
    for (int j = 0; j < 8; ++j) {
        const int trow = j + 8 * g;            // C/D layout: VGPR j -> row j (+8 for hi half)
        float pr[4];
        float like = 0.0f;
#pragma unroll
        for (int mt = 0; mt < 4; ++mt) {
            pr[mt] = expf(-0.5f * acc[mt][j] + s_m[mt]);
            like += pr[mt];
        }
        like += __shfl_xor(like, 1);
        like += __shfl_xor(like, 2);
        like += __shfl_xor(like, 4);
        like += __shfl_xor(like, 8);           // sum over m within 16-lane group
        const float inv = 1.0f / like;
#pragma unroll
        for (int mt = 0; mt < 4; ++mt)
            myLds[trow * 64 + mt * 16 + n] = pr[mt] * inv;
        if (n == 0) ll_acc += logf(like);      // one contribution per t row
    }
    __syncthreads();

    // Stage 3: e_means = posteriors @ means  (16x64 @ 64x16, 16 K-steps)
    v8f acc2 = {};
#pragma unroll
    for (int kk = 0; kk < 16; ++kk) {
        const int k = kk * 4;
        v2f a, b;
        a.x = myLds[n * 64 + k + 2 * g];
        a.y = myLds[n * 64 + k + 1 + 2 * g];
        b.x = means[(k + 2 * g) * NDIM + n];
        b.y = means[(k + 1 + 2 * g) * NDIM + n];
        acc2 = __builtin_amdgcn_wmma_f32_16x16x4_f32(false, a, false, b,
                                                     (short)0, acc2, false, false);
    }
#pragma unroll
    for (int j = 0; j < 8; ++j) {
        const int t = t_base + j + 8 * g;
        e_means[t * NDIM + n] = acc2[j];
    }

    // log-likelihood partial sum (lanes 0 and 16 hold data; full-wave reduce)
    float v = ll_acc;
    v += __shfl_xor(v, 1);
    v += __shfl_xor(v, 2);
    v += __shfl_xor(v, 4);
    v += __shfl_xor(v, 8);
    v += __shfl_xor(v, 16);
    if (lane == 0) wave_sums[wave] = v;
    __syncthreads();
    if (tid == 0) {
        float s = 0.0f;
        for (int w = 0; w < 8; ++w) s += wave_sums[w];
        partials[blockIdx.x] = s;
    }
}

// ---------------- Deterministic fixed-order final reduction
__global__ void gmm_reduce(const float* __restrict__ partials, int npart,
                           float invT, float* __restrict__ out) {
    __shared__ float sm[256];
    float s = 0.0f;
    for (int i = threadIdx.x; i < npart; i += 256) s += partials[i];
    sm[threadIdx.x] = s;
    __syncthreads();
    for (int off = 128; off > 0; off >>= 1) {
        if ((int)threadIdx.x < off) sm[threadIdx.x] += sm[threadIdx.x + off];
        __syncthreads();
    }
    if (threadIdx.x == 0) out[0] = sm[0] * invT;
}

extern "C" void kernel_launch(void* const* d_in, const int* in_sizes, int n_in,
                              void* d_out, int out_size, void* d_ws, size_t ws_size,
                              hipStream_t stream) {
    const float* data  = (const float*)d_in[0];
    const float* wghts = (const float*)d_in[1];
    const float* means = (const float*)d_in[2];
    const float* dcovs = (const float*)d_in[3];
    float* out = (float*)d_out;     // out[0] = log_likes, out[1..] = e_means (T,16)
    float* ws  = (float*)d_ws;

    float* Pws      = ws;           // 64*16
    float* Qws      = ws + 1024;    // 64*16
    float* Sws      = ws + 2048;    // 64
    float* partials = ws + 2112;    // nblocks

    const int T = in_sizes[0] / NDIM;       // 262144
    const int nblocks = T / 128;            // 8 waves * 16 t-rows per block

    gmm_precompute<<<1, 64, 0, stream>>>(wghts, means, dcovs, Pws, Qws, Sws);
    gmm_main<<<nblocks, 256, 0, stream>>>(data, means, Pws, Qws, Sws,
                                          out + 1, partials);
    gmm_reduce<<<1, 256, 0, stream>>>(partials, nblocks, 1.0f / (float)T, out);
}